// BackboneModule_69604239999490
// MI455X (gfx1250) — compile-verified
//
#include <hip/hip_runtime.h>
#include <hip/hip_bf16.h>

// Backbone/torsion rigid-frame atom placement (AlphaFold-style), MI455X gfx1250.
// Memory-bound (≈185 MB traffic @ 23.3 TB/s ≈ 8 us floor; ~0.6 GFLOP total).
// No GEMM structure -> WMMA inapplicable; CDNA5 paths used: async global->LDS
// table copies (ASYNCcnt), bank-conflict-free wave32 LDS layout, global prefetch.

#define NRES_T 21
#define NRIG   8
#define NATM   24
#define TPB    128

#if defined(__HIP_DEVICE_COMPILE__) && __has_builtin(__builtin_amdgcn_global_load_async_to_lds_b32)
#define USE_ASYNC_LDS 1
#else
#define USE_ASYNC_LDS 0
#endif

#if USE_ASYNC_LDS
typedef __attribute__((address_space(1))) int* gptr_i32;
typedef __attribute__((address_space(3))) int* lptr_i32;
#endif

__device__ __forceinline__ void lds_copy_f32(float* dst, const float* src, int n, int tid) {
#if USE_ASYNC_LDS
  for (int i = tid; i < n; i += TPB) {
    __builtin_amdgcn_global_load_async_to_lds_b32(
        (gptr_i32)(void*)(src + i),
        (lptr_i32)(void*)(dst + i),
        0, 0);
  }
#else
  for (int i = tid; i < n; i += TPB) dst[i] = src[i];
#endif
}

__device__ __forceinline__ void lds_copy_wait() {
#if USE_ASYNC_LDS
#if __has_builtin(__builtin_amdgcn_s_wait_asynccnt)
  __builtin_amdgcn_s_wait_asynccnt(0);
#else
  asm volatile("s_wait_asynccnt 0" ::: "memory");
#endif
#endif
}

__global__ void __launch_bounds__(TPB)
backbone_place_atoms(const float* __restrict__ bb,          // (N,4,3)
                     const float* __restrict__ sc,          // (N,7,2)
                     const float* __restrict__ pos0,        // (N,3)
                     const float* __restrict__ g_transforms,// (21,8,4,3)
                     const float* __restrict__ g_rigids,    // (21,24,3)
                     const int*   __restrict__ g_restype,   // (N)
                     const int*   __restrict__ g_tdep,      // (21,8)
                     const int*   __restrict__ g_rdep,      // (21,24)
                     float* __restrict__ outR,              // (N,24,3)
                     float* __restrict__ outF,              // (N,4,3)
                     int N)
{
  __shared__ float s_tf[NRES_T * NRIG * 12];   // 8064 B
  __shared__ float s_rg[NRES_T * NATM * 3];    // 6048 B
  __shared__ short s_td[NRES_T * NRIG];        //  336 B
  __shared__ short s_rd[NRES_T * NATM];        // 1008 B
  // Transposed per-thread frame store: element k of thread tid at [k*TPB + tid].
  // Lane stride = 1 DWORD -> conflict-free even with divergent frame indices
  // (index term is a multiple of 128 DWORDs == 0 mod 64 banks).
  __shared__ float s_opr[NRIG * 12 * TPB];     // 49152 B   (total 64608 B)

  const int tid = threadIdx.x;

  lds_copy_f32(s_tf, g_transforms, NRES_T * NRIG * 12, tid);
  lds_copy_f32(s_rg, g_rigids,     NRES_T * NATM * 3,  tid);
  for (int i = tid; i < NRES_T * NRIG; i += TPB) s_td[i] = (short)g_tdep[i];
  for (int i = tid; i < NRES_T * NATM; i += TPB) s_rd[i] = (short)g_rdep[i];
  lds_copy_wait();
  __syncthreads();

  const int r = blockIdx.x * TPB + tid;
  if (r >= N) return;

  __builtin_prefetch(bb + (size_t)r * 12, 0, 0);  // global_prefetch_b8
  __builtin_prefetch(sc + (size_t)r * 14, 0, 0);

  const int t = g_restype[r];

  // All 8 composed frames kept in registers. Layout per frame: rows 0..2 of the
  // 3x3 rotation at [0..8], translation at [9..11] (matches numpy (4,3) rows).
  float O[NRIG * 12];

  // ---- frame 0: transforms[t][0] o [bb_rot | bb_trans + pos0]
  {
    const float4* bb4 = (const float4*)bb;          // 48B/residue, 16B aligned
    float4 b0 = bb4[(size_t)r * 3 + 0];
    float4 b1 = bb4[(size_t)r * 3 + 1];
    float4 b2 = bb4[(size_t)r * 3 + 2];
    float B[12] = {b0.x, b0.y, b0.z, b0.w, b1.x, b1.y, b1.z, b1.w,
                   b2.x, b2.y, b2.z, b2.w};
    B[9]  += pos0[(size_t)r * 3 + 0];
    B[10] += pos0[(size_t)r * 3 + 1];
    B[11] += pos0[(size_t)r * 3 + 2];
    const float* T = &s_tf[(t * NRIG + 0) * 12];
#pragma unroll
    for (int i = 0; i < 3; ++i) {
      const float ti0 = T[i * 3 + 0], ti1 = T[i * 3 + 1], ti2 = T[i * 3 + 2];
#pragma unroll
      for (int k = 0; k < 3; ++k)
        O[i * 3 + k] = ti0 * B[0 + k] + ti1 * B[3 + k] + ti2 * B[6 + k];
      O[9 + i] = ti0 * B[9] + ti1 * B[10] + ti2 * B[11] + T[9 + i];
    }
  }

  // ---- frames 1..7: transforms[t][g] o Rx(cos,sin)   (Rx trans == 0)
  {
    const float2* sc2 = (const float2*)sc;          // 8B aligned per residue
#pragma unroll
    for (int g = 1; g < NRIG; ++g) {
      const float2 cs = sc2[(size_t)r * 7 + (g - 1)];  // .x = cos, .y = sin
      const float* T = &s_tf[(t * NRIG + g) * 12];
#pragma unroll
      for (int i = 0; i < 3; ++i) {
        const float t1 = T[i * 3 + 1], t2 = T[i * 3 + 2];
        O[g * 12 + i * 3 + 0] = T[i * 3 + 0];
        O[g * 12 + i * 3 + 1] =  cs.x * t1 + cs.y * t2;
        O[g * 12 + i * 3 + 2] = -cs.y * t1 + cs.x * t2;
      }
      O[g * 12 + 9]  = T[9];
      O[g * 12 + 10] = T[10];
      O[g * 12 + 11] = T[11];
    }
  }

  // ---- dependency chain: O[i] = combine(O[dep[i]], O[i]), dep[i] < i.
  // Parent selected from registers via v_cndmask trees (no LDS round-trips).
#pragma unroll
  for (int i = 1; i < NRIG; ++i) {
    const int d = (int)s_td[t * NRIG + i];
    float P[12];
#pragma unroll
    for (int k = 0; k < 12; ++k) P[k] = O[k];        // default: parent 0
#pragma unroll
    for (int j = 1; j < NRIG; ++j) {
      if (j < i) {                                   // statically pruned
        const bool m = (d == j);
#pragma unroll
        for (int k = 0; k < 12; ++k) P[k] = m ? O[j * 12 + k] : P[k];
      }
    }
    float Rm[12];
#pragma unroll
    for (int a = 0; a < 3; ++a) {
      const float p0 = P[a * 3 + 0], p1 = P[a * 3 + 1], p2 = P[a * 3 + 2];
#pragma unroll
      for (int b = 0; b < 3; ++b)
        Rm[a * 3 + b] = p0 * O[i * 12 + b] + p1 * O[i * 12 + 3 + b] +
                        p2 * O[i * 12 + 6 + b];
      Rm[9 + a] = p0 * O[i * 12 + 9] + p1 * O[i * 12 + 10] +
                  p2 * O[i * 12 + 11] + P[9 + a];
    }
#pragma unroll
    for (int k = 0; k < 12; ++k) O[i * 12 + k] = Rm[k];
  }

  // ---- spill final frames to LDS once (dynamic per-atom indexing below)
#pragma unroll
  for (int k = 0; k < NRIG * 12; ++k)
    s_opr[k * TPB + tid] = O[k];

  // ---- atom placement: R[a] = M_rot(dep[a]) @ rigids[a] + M_trans(dep[a])
  float* outp = outR + (size_t)r * (NATM * 3);       // 72 contiguous floats
#pragma unroll
  for (int a = 0; a < NATM; ++a) {
    const int d = (int)s_rd[t * NATM + a];
    const float* v = &s_rg[(t * NATM + a) * 3];
    const float vx = v[0], vy = v[1], vz = v[2];
    float M[12];
#pragma unroll
    for (int k = 0; k < 12; ++k) M[k] = s_opr[(d * 12 + k) * TPB + tid];
    outp[a * 3 + 0] = M[0] * vx + M[1] * vy + M[2] * vz + M[9];
    outp[a * 3 + 1] = M[3] * vx + M[4] * vy + M[5] * vz + M[10];
    outp[a * 3 + 2] = M[6] * vx + M[7] * vy + M[8] * vz + M[11];
  }

  // ---- second output: frame used by atom 0 (opr[rigids_dep[t][0]])
  {
    const int d0 = (int)s_rd[t * NATM + 0];
    float F[12];
#pragma unroll
    for (int k = 0; k < 12; ++k) F[k] = s_opr[(d0 * 12 + k) * TPB + tid];
    float4* f4 = (float4*)(outF + (size_t)r * 12);   // 48B, 16B aligned
    f4[0] = make_float4(F[0], F[1], F[2],  F[3]);
    f4[1] = make_float4(F[4], F[5], F[6],  F[7]);
    f4[2] = make_float4(F[8], F[9], F[10], F[11]);
  }
}

extern "C" void kernel_launch(void* const* d_in, const int* in_sizes, int n_in,
                              void* d_out, int out_size, void* d_ws, size_t ws_size,
                              hipStream_t stream) {
  const float* bb       = (const float*)d_in[0];
  const float* sc       = (const float*)d_in[1];
  const float* pos0     = (const float*)d_in[2];
  const float* tf_tab   = (const float*)d_in[3];
  const float* rg_tab   = (const float*)d_in[4];
  const int*   restype  = (const int*)d_in[5];
  const int*   tdep_tab = (const int*)d_in[6];
  const int*   rdep_tab = (const int*)d_in[7];

  const int N = in_sizes[5];                 // residue_type flat count
  float* outR = (float*)d_out;               // (N,24,3)
  float* outF = outR + (size_t)N * NATM * 3; // (N,4,3), concatenated flat

  const int blocks = (N + TPB - 1) / TPB;
  backbone_place_atoms<<<blocks, TPB, 0, stream>>>(
      bb, sc, pos0, tf_tab, rg_tab, restype, tdep_tab, rdep_tab,
      outR, outF, N);
}